// VisitEncoder_69045894250729
// MI455X (gfx1250) — compile-verified
//
#include <hip/hip_runtime.h>
#include <cstdint>

#define LCODES 48
#define DIM_ 64
#define WAVES_PER_BLOCK 8
#define BLOCK_THREADS (WAVES_PER_BLOCK * 32)

typedef unsigned int u32x4 __attribute__((ext_vector_type(4)));
typedef int i32x8 __attribute__((ext_vector_type(8)));
typedef int i32x4 __attribute__((ext_vector_type(4)));

#ifndef USE_TDM
#if __has_builtin(__builtin_amdgcn_tensor_load_to_lds)
#define USE_TDM 1
#else
#define USE_TDM 0
#endif
#endif

// Per-row Poincare logmap0 scale: atanh(min(||x||,1-eps)) / max(||x||,1e-15)
__device__ __forceinline__ float row_scale(float sq) {
  float norm = __builtin_amdgcn_sqrtf(sq);
  norm = fmaxf(norm, 1e-15f);                 // MIN_NORM
  float arg = fminf(norm, 1.0f - 1e-7f);      // ATANH clamp (sqrt_c = 1)
  // atanh(a) = 0.5 * ln((1+a)/(1-a))
  float ratio = (1.0f + arg) * __builtin_amdgcn_rcpf(1.0f - arg);
  return 0.5f * __logf(ratio) * __builtin_amdgcn_rcpf(norm);
}

// Pass 1: per table row compute scale; write scale*row (table mode) or the
// scalar scale (vector mode). Half-wave (16 lanes x float4) per row.
__global__ __launch_bounds__(BLOCK_THREADS) void prescale_kernel(
    const float* __restrict__ emb, float* __restrict__ dst,
    int numCodes, int writeTable)
{
  const int wave = threadIdx.x >> 5;
  const int lane = threadIdx.x & 31;
  const int half = lane >> 4;
  const int sub  = lane & 15;
  const int row = blockIdx.x * (WAVES_PER_BLOCK * 2) + wave * 2 + half;
  if (row >= numCodes) return;

  const float4* __restrict__ emb4 = (const float4*)emb;
  float4 x = emb4[(size_t)row * (DIM_ / 4) + sub];

  float sq = x.x * x.x;
  sq = fmaf(x.y, x.y, sq);
  sq = fmaf(x.z, x.z, sq);
  sq = fmaf(x.w, x.w, sq);
  sq += __shfl_xor(sq, 1, 32);
  sq += __shfl_xor(sq, 2, 32);
  sq += __shfl_xor(sq, 4, 32);
  sq += __shfl_xor(sq, 8, 32);

  const float s = row_scale(sq);
  if (writeTable) {
    float4 r;
    r.x = s * x.x; r.y = s * x.y; r.z = s * x.z; r.w = s * x.w;
    ((float4*)dst)[(size_t)row * (DIM_ / 4) + sub] = r;
  } else if (sub == 0) {
    dst[row] = s;
  }
}

// Pass 2: masked-mean gather. MODE 2: src is prescaled table (pure gather-sum).
// MODE 1: src is raw emb + wscale vector. MODE 0: fused per-row math.
template <int MODE>
__global__ __launch_bounds__(BLOCK_THREADS) void visit_gather_kernel(
    const int* __restrict__ code_ids,
    const float* __restrict__ src,
    const float* __restrict__ wscale,
    float* __restrict__ out,
    int N)
{
  __shared__ __align__(16) int sIds[WAVES_PER_BLOCK * LCODES]; // 384 ints

  const int nBase = blockIdx.x * WAVES_PER_BLOCK;
  int navail = N - nBase;
  if (navail > WAVES_PER_BLOCK) navail = WAVES_PER_BLOCK;
  const int nints = navail * LCODES;

#if USE_TDM
  // TDM: DMA the block's code-id tile (nints x 1, 4B elems) into LDS.
  // Descriptor per CDNA5 ISA 8.3/8.4; wave 0 issues, waits TENSORcnt.
  if (threadIdx.x < 32) {
    unsigned long long gaddr =
        (unsigned long long)(uintptr_t)(code_ids + (size_t)nBase * LCODES);
    unsigned lds_off = (unsigned)(uintptr_t)(&sIds[0]);
    u32x4 g0;
    g0.x = 1u;                                         // count=1 (valid)
    g0.y = lds_off;                                    // lds_addr
    g0.z = (unsigned)(gaddr & 0xFFFFFFFFull);          // global_addr[31:0]
    g0.w = (unsigned)((gaddr >> 32) & 0x01FFFFFFull)   // global_addr[56:32]
         | 0x80000000u;                                // type=2 ("image")
    i32x8 g1;
    g1[0] = 0x00020000;                                // wg_mask=0, data_size=4B
    g1[1] = (nints & 0xFFFF) << 16;                    // tensor_dim0 lo16
    g1[2] = (nints >> 16) | (1 << 16);                 // tensor_dim0 hi | tensor_dim1=1
    g1[3] = (nints & 0xFFFF) << 16;                    // tile_dim0 = nints
    g1[4] = 1;                                         // tile_dim1=1, tile_dim2=0
    g1[5] = nints;                                     // tensor_dim0_stride lo32
    g1[6] = 0;
    g1[7] = 0;
    i32x4 gz4 = {0, 0, 0, 0};
    i32x8 gz8 = {0, 0, 0, 0, 0, 0, 0, 0};
    __builtin_amdgcn_tensor_load_to_lds(g0, g1, gz4, gz4, gz8, 0);
    __builtin_amdgcn_s_wait_tensorcnt(0);
  }
#else
  {
    const int* gsrc = code_ids + (size_t)nBase * LCODES;
    for (int i = threadIdx.x; i < nints; i += BLOCK_THREADS) sIds[i] = gsrc[i];
  }
#endif
  __syncthreads();

  const int wave = threadIdx.x >> 5;
  const int lane = threadIdx.x & 31;
  const int half = lane >> 4;   // which code of the pair
  const int sub  = lane & 15;   // 4 dims/lane -> 16 lanes cover D=64
  const int n = nBase + wave;
  if (n >= N) return;

  const float4* __restrict__ src4 = (const float4*)src;
  const int* wIds = sIds + wave * LCODES;

  float4 acc = make_float4(0.f, 0.f, 0.f, 0.f);
  float cnt = 0.f;

  // software-pipelined gather: b128 per lane (256B/row per half-wave)
  int idc = wIds[half];
  int safec = idc >= 0 ? idc : 0;
  float4 x = src4[(size_t)safec * (DIM_ / 4) + sub];
  float sc = (MODE == 1) ? wscale[safec] : 0.f;

  for (int j = 0; j < LCODES; j += 2) {
    int idn = -1;
    float4 xn = make_float4(0.f, 0.f, 0.f, 0.f);
    float sn = 0.f;
    if (j + 2 < LCODES) {
      idn = wIds[j + 2 + half];
      int safen = idn >= 0 ? idn : 0;
      xn = src4[(size_t)safen * (DIM_ / 4) + sub];
      if (MODE == 1) sn = wscale[safen];
    }

    const float vmask = (idc >= 0) ? 1.0f : 0.0f;
    float w;
    if (MODE == 0) {
      float sq = x.x * x.x;
      sq = fmaf(x.y, x.y, sq);
      sq = fmaf(x.z, x.z, sq);
      sq = fmaf(x.w, x.w, sq);
      sq += __shfl_xor(sq, 1, 32);
      sq += __shfl_xor(sq, 2, 32);
      sq += __shfl_xor(sq, 4, 32);
      sq += __shfl_xor(sq, 8, 32);
      w = vmask * row_scale(sq);
    } else if (MODE == 1) {
      w = vmask * sc;
    } else {
      w = vmask;            // rows already prescaled
    }

    acc.x = fmaf(w, x.x, acc.x);
    acc.y = fmaf(w, x.y, acc.y);
    acc.z = fmaf(w, x.z, acc.z);
    acc.w = fmaf(w, x.w, acc.w);
    cnt += vmask;

    idc = idn;
    x = xn;
    sc = sn;
  }

  // combine the two half-wave partial sums (even codes + odd codes)
  acc.x += __shfl_xor(acc.x, 16, 32);
  acc.y += __shfl_xor(acc.y, 16, 32);
  acc.z += __shfl_xor(acc.z, 16, 32);
  acc.w += __shfl_xor(acc.w, 16, 32);
  cnt   += __shfl_xor(cnt,   16, 32);

  if (half == 0) {
    const float inv = __builtin_amdgcn_rcpf(fmaxf(cnt, 1.0f));
    float4 r;
    r.x = acc.x * inv;
    r.y = acc.y * inv;
    r.z = acc.z * inv;
    r.w = acc.w * inv;
    ((float4*)out)[(size_t)n * (DIM_ / 4) + sub] = r;
  }
}

extern "C" void kernel_launch(void* const* d_in, const int* in_sizes, int n_in,
                              void* d_out, int out_size, void* d_ws, size_t ws_size,
                              hipStream_t stream) {
  const int* code_ids = (const int*)d_in[0];
  const float* emb    = (const float*)d_in[1];
  float* out          = (float*)d_out;
  const int N = in_sizes[0] / LCODES;
  const int numCodes = in_sizes[1] / DIM_;
  const int vblocks = (N + WAVES_PER_BLOCK - 1) / WAVES_PER_BLOCK;
  const int pblocks = (numCodes + WAVES_PER_BLOCK * 2 - 1) / (WAVES_PER_BLOCK * 2);

  const size_t tabBytes = (size_t)numCodes * DIM_ * sizeof(float);
  const size_t vecBytes = (size_t)numCodes * sizeof(float);

  if (ws_size >= tabBytes) {
    float* scaledTab = (float*)d_ws;
    prescale_kernel<<<pblocks, BLOCK_THREADS, 0, stream>>>(emb, scaledTab, numCodes, 1);
    visit_gather_kernel<2><<<vblocks, BLOCK_THREADS, 0, stream>>>(
        code_ids, scaledTab, nullptr, out, N);
  } else if (ws_size >= vecBytes) {
    float* wscale = (float*)d_ws;
    prescale_kernel<<<pblocks, BLOCK_THREADS, 0, stream>>>(emb, wscale, numCodes, 0);
    visit_gather_kernel<1><<<vblocks, BLOCK_THREADS, 0, stream>>>(
        code_ids, emb, wscale, out, N);
  } else {
    visit_gather_kernel<0><<<vblocks, BLOCK_THREADS, 0, stream>>>(
        code_ids, emb, nullptr, out, N);
  }
  (void)n_in; (void)out_size;
}